// Multi_Head_Attention_17789754540274
// MI455X (gfx1250) — compile-verified
//
#include <hip/hip_runtime.h>

// MHA: B=4, S=2048, D=1024, H=16, depth=64, fp32. Outputs: out [B,S,D] then attn [B,H,S,S].
#define BB 4
#define SS 2048
#define DD 1024
#define HH 16
#define DEP 64

typedef float v2f __attribute__((ext_vector_type(2)));
typedef float v4f __attribute__((ext_vector_type(4)));
typedef float v8f __attribute__((ext_vector_type(8)));
typedef int v4i __attribute__((ext_vector_type(4)));

#define AS1 __attribute__((address_space(1)))
#define AS3 __attribute__((address_space(3)))

#if defined(__has_builtin)
#if __has_builtin(__builtin_amdgcn_global_load_async_to_lds_b128)
#define HAVE_ASYNC_LDS 1
#endif
#endif
#ifndef HAVE_ASYNC_LDS
#define HAVE_ASYNC_LDS 0
#endif

// D = A(16x4 f32) * B(4x16 f32) + C(16x16 f32), wave32 WMMA.
__device__ __forceinline__ v8f wmma_f32(v2f a, v2f b, v8f c) {
  return __builtin_amdgcn_wmma_f32_16x16x4_f32(
      /*neg_a=*/false, a, /*neg_b=*/false, b,
      /*c_mod=*/(short)0, c, /*reuse_a=*/false, /*reuse_b=*/false);
}

// 16B global -> LDS copy; async (ASYNCcnt) when the toolchain exposes it.
__device__ __forceinline__ void lds_copy16(const float* gsrc, float* ldst) {
#if HAVE_ASYNC_LDS
  __builtin_amdgcn_global_load_async_to_lds_b128(
      (AS1 v4i*)gsrc, (AS3 v4i*)ldst, /*offset=*/0, /*cpol=*/0);
#else
  *(v4f*)ldst = *(const v4f*)gsrc;
#endif
}

// Wait for per-wave async copies, then workgroup barrier.
__device__ __forceinline__ void lds_stage_fence() {
#if HAVE_ASYNC_LDS
#if __has_builtin(__builtin_amdgcn_s_wait_asynccnt)
  __builtin_amdgcn_s_wait_asynccnt(0);
#else
  asm volatile("s_wait_asynccnt 0" ::: "memory");
#endif
#endif
  __syncthreads();
}

// ---------------------------------------------------------------------------
// GEMM: Y = X[8192x1024] @ W[1024x1024] + bias.  MODE 0: Y[m*D+n] linear.
// MODE 1: head-split scatter Y[((b*H+h)*S+s)*64+d].
// Block = 256 thr = 8 waves; block tile 128x64; wave tile 16x64 (4 C frags).
// W tiles (32x64) staged through LDS (async when available), shared by 8 waves.
// ---------------------------------------------------------------------------
template <int MODE>
__global__ __launch_bounds__(256) void gemm_bias_kernel(
    const float* __restrict__ X, const float* __restrict__ W,
    const float* __restrict__ bias, float* __restrict__ Y) {
  __shared__ __align__(16) float ldsW[32 * 64];  // 8 KB
  const int w = threadIdx.x >> 5;
  const int lane = threadIdx.x & 31;
  const int l15 = lane & 15;
  const int koff = (lane >> 4) * 2;  // K-pair selected by lane half
  const int n0 = blockIdx.x * 64;
  const int m0 = blockIdx.y * 128 + w * 16;
  const int arow = (m0 + l15) * DD;

  v8f acc[4] = {};

  for (int kk = 0; kk < DD; kk += 32) {
    // Stage W[kk..kk+31][n0..n0+63] (2048 floats) : 8 floats per thread.
    {
      const int base = threadIdx.x * 8;  // multiple of 8, row-aligned
      const int r0 = base >> 6, c0 = base & 63;
      const float* src = &W[(size_t)(kk + r0) * DD + n0 + c0];
      lds_copy16(&src[0], &ldsW[base]);
      lds_copy16(&src[4], &ldsW[base + 4]);
    }
    lds_stage_fence();
    if (kk + 32 < DD)  // hint next A chunk into cache (global_prefetch_b8)
      __builtin_prefetch(&X[arow + kk + 32], 0, 0);
#pragma unroll
    for (int j = 0; j < 8; ++j) {
      const int k4 = j * 4;
      v2f a = *(const v2f*)&X[arow + kk + k4 + koff];
#pragma unroll
      for (int t = 0; t < 4; ++t) {
        const int col = t * 16 + l15;
        v2f b;
        b.x = ldsW[(k4 + koff) * 64 + col];
        b.y = ldsW[(k4 + koff + 1) * 64 + col];
        acc[t] = wmma_f32(a, b, acc[t]);
      }
    }
    __syncthreads();
  }

#pragma unroll
  for (int t = 0; t < 4; ++t) {
    const int n = n0 + t * 16 + l15;
    const float bv = bias[n];
#pragma unroll
    for (int r = 0; r < 8; ++r) {
      const int m = m0 + r + ((lane >> 4) * 8);
      const float val = acc[t][r] + bv;
      if (MODE == 0) {
        Y[(size_t)m * DD + n] = val;
      } else {
        const int b = m / SS, s = m % SS, h = n >> 6, d = n & 63;
        Y[(((size_t)(b * HH + h) * SS) + s) * DEP + d] = val;
      }
    }
  }
}

// ---------------------------------------------------------------------------
// scores = (q @ k^T) * 1/sqrt(64) + mask * -1e9, written pre-softmax to attn.
// q,k: [BH][S][64]. Wave tile 16x64 of the S x S score matrix.
// Scores are a 1.07GB stream (>> 192MB L2): store non-temporal.
// ---------------------------------------------------------------------------
__global__ __launch_bounds__(256) void scores_kernel(
    const float* __restrict__ q, const float* __restrict__ k,
    const float* __restrict__ Mask, float* __restrict__ attn) {
  const int w = threadIdx.x >> 5;
  const int lane = threadIdx.x & 31;
  const int l15 = lane & 15;
  const int koff = (lane >> 4) * 2;
  const int bh = blockIdx.z;
  const int b = bh / HH;
  const int m0 = blockIdx.y * 16;
  const int n0 = blockIdx.x * 512 + w * 64;
  const float* qh = q + (size_t)bh * SS * DEP;
  const float* kh = k + (size_t)bh * SS * DEP;
  const int arow = (m0 + l15) * DEP;

  v8f acc[4] = {};
#pragma unroll
  for (int kk = 0; kk < DEP; kk += 4) {
    v2f a = *(const v2f*)&qh[arow + kk + koff];
#pragma unroll
    for (int t = 0; t < 4; ++t) {
      const int n = n0 + t * 16 + l15;
      v2f bf = *(const v2f*)&kh[(size_t)n * DEP + kk + koff];  // k^T fragment
      acc[t] = wmma_f32(a, bf, acc[t]);
    }
  }

  const float scale = 0.125f;  // 1/sqrt(64)
  float* oh = attn + (size_t)bh * SS * SS;
  const float* mh = Mask + (size_t)b * SS * SS;
#pragma unroll
  for (int t = 0; t < 4; ++t) {
    const int n = n0 + t * 16 + l15;
#pragma unroll
    for (int r = 0; r < 8; ++r) {
      const int m = m0 + r + ((lane >> 4) * 8);
      const float sc = acc[t][r] * scale + mh[(size_t)m * SS + n] * (-1e9f);
      __builtin_nontemporal_store(sc, &oh[(size_t)m * SS + n]);
    }
  }
}

// ---------------------------------------------------------------------------
// Row softmax over 2048 elems, in place. One wave32 per row; 64 floats/lane
// held in registers; wave reductions via shfl_xor. Pure 2.1GB stream: NT.
// ---------------------------------------------------------------------------
__global__ __launch_bounds__(256) void softmax_kernel(float* __restrict__ attn) {
  const int w = threadIdx.x >> 5;
  const int lane = threadIdx.x & 31;
  float* row = attn + ((size_t)blockIdx.x * 8 + w) * SS;

  v4f vals[16];
  float mx = -3.0e38f;
#pragma unroll
  for (int i = 0; i < 16; ++i) {
    vals[i] = __builtin_nontemporal_load((const v4f*)&row[i * 128 + lane * 4]);
    mx = fmaxf(mx, fmaxf(fmaxf(vals[i].x, vals[i].y), fmaxf(vals[i].z, vals[i].w)));
  }
#pragma unroll
  for (int off = 16; off > 0; off >>= 1) mx = fmaxf(mx, __shfl_xor(mx, off, 32));

  float sum = 0.0f;
#pragma unroll
  for (int i = 0; i < 16; ++i) {
    vals[i].x = __expf(vals[i].x - mx);
    vals[i].y = __expf(vals[i].y - mx);
    vals[i].z = __expf(vals[i].z - mx);
    vals[i].w = __expf(vals[i].w - mx);
    sum += vals[i].x + vals[i].y + vals[i].z + vals[i].w;
  }
#pragma unroll
  for (int off = 16; off > 0; off >>= 1) sum += __shfl_xor(sum, off, 32);
  const float inv = 1.0f / sum;

#pragma unroll
  for (int i = 0; i < 16; ++i) {
    vals[i] *= inv;
    __builtin_nontemporal_store(vals[i], (v4f*)&row[i * 128 + lane * 4]);
  }
}

// ---------------------------------------------------------------------------
// rep = attn @ v  (per head: [S x S] @ [S x 64]); result into concat layout
// concat[(b*S+m)*D + h*64 + n].  V staged through LDS in 128x64 chunks
// (async copies), attn A-fragments read non-temporal (1.07GB stream).
// ---------------------------------------------------------------------------
__global__ __launch_bounds__(256) void attnv_kernel(
    const float* __restrict__ attn, const float* __restrict__ v,
    float* __restrict__ concat) {
  __shared__ __align__(16) float ldsV[128 * 64];  // 32 KB
  const int w = threadIdx.x >> 5;
  const int lane = threadIdx.x & 31;
  const int l15 = lane & 15;
  const int koff = (lane >> 4) * 2;
  const int bh = blockIdx.y;
  const int b = bh / HH, h = bh % HH;
  const int m0 = blockIdx.x * 128 + w * 16;
  const float* ah = attn + (size_t)bh * SS * SS;
  const float* vh = v + (size_t)bh * SS * DEP;
  const size_t arow = (size_t)(m0 + l15) * SS;

  v8f acc[4] = {};
  for (int kc = 0; kc < SS; kc += 128) {
    // Stage v[kc..kc+127][0..63]: 8192 floats, 8 x 16B per thread.
#pragma unroll
    for (int i = 0; i < 8; ++i) {
      const int f = threadIdx.x + i * 256;  // float4 index 0..2047
      lds_copy16(&vh[(size_t)kc * DEP + f * 4], &ldsV[f * 4]);
    }
    lds_stage_fence();
#pragma unroll 4
    for (int j = 0; j < 32; ++j) {
      const int k4 = j * 4;
      v2f a = __builtin_nontemporal_load(
          (const v2f*)&ah[arow + kc + k4 + koff]);
#pragma unroll
      for (int t = 0; t < 4; ++t) {
        const int col = t * 16 + l15;
        v2f bf;
        bf.x = ldsV[(k4 + koff) * 64 + col];
        bf.y = ldsV[(k4 + koff + 1) * 64 + col];
        acc[t] = wmma_f32(a, bf, acc[t]);
      }
    }
    __syncthreads();
  }

#pragma unroll
  for (int t = 0; t < 4; ++t) {
    const int n = t * 16 + l15;
#pragma unroll
    for (int r = 0; r < 8; ++r) {
      const int m = m0 + r + ((lane >> 4) * 8);
      concat[(size_t)(b * SS + m) * DD + h * DEP + n] = acc[t][r];
    }
  }
}

// ---------------------------------------------------------------------------
extern "C" void kernel_launch(void* const* d_in, const int* in_sizes, int n_in,
                              void* d_out, int out_size, void* d_ws,
                              size_t ws_size, hipStream_t stream) {
  const float* Q = (const float*)d_in[0];
  const float* K = (const float*)d_in[1];
  const float* V = (const float*)d_in[2];
  const float* Mask = (const float*)d_in[3];
  const float* Wq = (const float*)d_in[4];
  const float* bq = (const float*)d_in[5];
  const float* Wk = (const float*)d_in[6];
  const float* bk = (const float*)d_in[7];
  const float* Wv = (const float*)d_in[8];
  const float* bv = (const float*)d_in[9];
  const float* Wo = (const float*)d_in[10];
  const float* bo = (const float*)d_in[11];

  float* out = (float*)d_out;                // [B,S,D]
  float* attn = out + (size_t)BB * SS * DD;  // [B,H,S,S]

  const size_t qkvElems = (size_t)BB * SS * DD;  // 8.39M floats each
  float* q = (float*)d_ws;                       // ws: 4 x 33.5MB = 128MB
  float* k = q + qkvElems;
  float* v = k + qkvElems;
  float* concat = v + qkvElems;

  dim3 blk(256);
  dim3 gProj(DD / 64, (BB * SS) / 128);  // 16 x 64 blocks

  gemm_bias_kernel<1><<<gProj, blk, 0, stream>>>(Q, Wq, bq, q);
  gemm_bias_kernel<1><<<gProj, blk, 0, stream>>>(K, Wk, bk, k);
  gemm_bias_kernel<1><<<gProj, blk, 0, stream>>>(V, Wv, bv, v);

  scores_kernel<<<dim3(4, 128, BB * HH), blk, 0, stream>>>(q, k, Mask, attn);
  softmax_kernel<<<dim3((BB * HH * SS) / 8), blk, 0, stream>>>(attn);
  attnv_kernel<<<dim3(SS / 128, BB * HH), blk, 0, stream>>>(attn, v, concat);

  gemm_bias_kernel<0><<<gProj, blk, 0, stream>>>(concat, Wo, bo, out);
}